// GA_38001870635192
// MI455X (gfx1250) — compile-verified
//
#include <hip/hip_runtime.h>

// Problem constants (from reference): x is (B=16, C=256, T=2048) fp32.
#define B_ 16
#define C_ 256
#define T_ 2048
#define K_TOP 256            // T/FACTOR = 2048/8

typedef __attribute__((ext_vector_type(2))) float v2f;
typedef __attribute__((ext_vector_type(8))) float v8f;

// D = A(16x4 f32) * B(4x16 f32) + C(16x16 f32), full fp32 matrix pipe.
// 8-arg pattern: (neg_a, A, neg_b, B, c_mod, C, reuse_a, reuse_b)
#define WMMA_F32(a, b, c) \
  __builtin_amdgcn_wmma_f32_16x16x4_f32(false, (a), false, (b), (short)0, (c), false, false)

// ---------------------------------------------------------------------------
// Kernel A: inv_norm[b*T+t] = 1 / max(sqrt(sum_c x[b,c,t]^2), 1e-12)
// ---------------------------------------------------------------------------
__global__ void norm_kernel(const float* __restrict__ x, float* __restrict__ invn) {
  const int idx = blockIdx.x * blockDim.x + threadIdx.x;   // 0 .. B*T-1
  const int b = idx / T_;
  const int t = idx % T_;
  const float* xp = x + (size_t)b * C_ * T_ + t;
  float s = 0.f;
#pragma unroll 8
  for (int c = 0; c < C_; ++c) s += xp[(size_t)c * T_] * xp[(size_t)c * T_];
  invn[idx] = 1.f / fmaxf(sqrtf(s), 1e-12f);
}

// ---------------------------------------------------------------------------
// Kernel B: sim[b,t,s] = invn[t]*invn[s] * sum_c x[b,c,t]*x[b,c,s]
// One wave computes a 32(t) x 64(s) tile with V_WMMA_F32_16X16X4_F32.
// ---------------------------------------------------------------------------
__global__ void gram_kernel(const float* __restrict__ x,
                            const float* __restrict__ invn,
                            float* __restrict__ sim) {
  const int lane = threadIdx.x & 31;
  const int wid  = blockIdx.x * (blockDim.x >> 5) + (threadIdx.x >> 5);
  // per batch: (T/32)=64 t-tiles x (T/64)=32 s-tiles
  const int b   = wid / (64 * 32);
  const int rem = wid % (64 * 32);
  const int t_base = (rem / 32) * 32;
  const int s_base = (rem % 32) * 64;

  const float* xb = x + (size_t)b * C_ * T_;
  const float* nb = invn + b * T_;

  const int row = lane & 15;          // A-row m / B-col n per ISA layout
  const int kk  = (lane >> 4) * 2;    // K pair handled by this lane half

  float invA[2], invB[4];
#pragma unroll
  for (int mt = 0; mt < 2; ++mt) invA[mt] = nb[t_base + mt * 16 + row];
#pragma unroll
  for (int nt = 0; nt < 4; ++nt) invB[nt] = nb[s_base + nt * 16 + row];

  v8f acc[2][4] = {};

  for (int k0 = 0; k0 < C_; k0 += 4) {
    const float* base = xb + (size_t)(k0 + kk) * T_;
    v2f a[2], bb[4];
#pragma unroll
    for (int mt = 0; mt < 2; ++mt) {
      const int t = t_base + mt * 16 + row;
      a[mt].x = base[t]      * invA[mt];     // (k, m)
      a[mt].y = base[T_ + t] * invA[mt];     // (k+1, m)
    }
#pragma unroll
    for (int nt = 0; nt < 4; ++nt) {
      const int s = s_base + nt * 16 + row;
      bb[nt].x = base[s];
      bb[nt].y = base[T_ + s];
    }
#pragma unroll
    for (int mt = 0; mt < 2; ++mt)
#pragma unroll
      for (int nt = 0; nt < 4; ++nt)
        acc[mt][nt] = WMMA_F32(a[mt], bb[nt], acc[mt][nt]);
  }

  float* sb = sim + (size_t)b * T_ * T_;
#pragma unroll
  for (int mt = 0; mt < 2; ++mt)
#pragma unroll
    for (int nt = 0; nt < 4; ++nt) {
      const int s = s_base + nt * 16 + row;
#pragma unroll
      for (int i = 0; i < 8; ++i) {
        const int t = t_base + mt * 16 + i + (lane >> 4) * 8;  // C/D row map
        sb[(size_t)t * T_ + s] = acc[mt][nt][i] * invB[nt];
      }
    }
}

// ---------------------------------------------------------------------------
// Kernel C: per row of sim, exact k-th largest via 4-pass radix select on
// monotone float->uint keys; then sum of kept elements. One block per row.
// ---------------------------------------------------------------------------
__global__ void topk_kernel(const float* __restrict__ sim,
                            float* __restrict__ thr_out,
                            float* __restrict__ isum_out) {
  const int row = blockIdx.x;          // 0 .. B*T-1
  const int tid = threadIdx.x;         // 0 .. 255
  const float* rp = sim + (size_t)row * T_;

  __shared__ unsigned int keys[T_];
  __shared__ unsigned int hist[256];
  __shared__ unsigned int sh_prefix;
  __shared__ unsigned int sh_k;
  __shared__ float partial[256];

#pragma unroll
  for (int i = 0; i < 8; ++i) {
    const int j = tid + i * 256;
    unsigned u = __float_as_uint(rp[j]);
    keys[j] = (u & 0x80000000u) ? ~u : (u | 0x80000000u);  // order-preserving
  }
  if (tid == 0) { sh_prefix = 0u; sh_k = K_TOP; }
  __syncthreads();

  for (int pass = 3; pass >= 0; --pass) {
    hist[tid] = 0u;
    __syncthreads();
    const unsigned prefix = sh_prefix;
    const int shift = pass * 8;
#pragma unroll
    for (int i = 0; i < 8; ++i) {
      const unsigned u = keys[tid + i * 256];
      const bool match = (pass == 3) || ((u >> (shift + 8)) == prefix);
      if (match) atomicAdd(&hist[(u >> shift) & 0xFFu], 1u);
    }
    __syncthreads();
    if (tid == 0) {
      unsigned k = sh_k, cum = 0u, digit = 0u;
      for (int bin = 255; bin >= 0; --bin) {
        const unsigned c = hist[bin];
        if (cum + c >= k) { digit = (unsigned)bin; k -= cum; break; }
        cum += c;
      }
      sh_k = k;
      sh_prefix = (prefix << 8) | digit;
    }
    __syncthreads();
  }

  const unsigned tkey = sh_prefix;     // sortable key of k-th largest
  float s = 0.f;
#pragma unroll
  for (int i = 0; i < 8; ++i) {
    const unsigned u = keys[tid + i * 256];
    if (u >= tkey) {
      const unsigned fb = (u & 0x80000000u) ? (u & 0x7FFFFFFFu) : ~u;
      s += __uint_as_float(fb);
    }
  }
  partial[tid] = s;
  __syncthreads();
  for (int off = 128; off > 0; off >>= 1) {
    if (tid < off) partial[tid] += partial[tid + off];
    __syncthreads();
  }
  if (tid == 0) {
    const unsigned fb = (tkey & 0x80000000u) ? (tkey & 0x7FFFFFFFu) : ~tkey;
    thr_out[row] = __uint_as_float(fb);
    float sum = partial[0];
    if (sum == 0.f) sum = 1.f;
    isum_out[row] = 1.f / sum;
  }
}

// ---------------------------------------------------------------------------
// Kernel D: out1[b,c,t] = x[b,c,t] + sum_s w[t,s]*x[b,c,s],
//           w = (sim >= thr[t]) ? sim * isum[t] : 0.
// One wave computes a 32(t) x 64(c) tile with V_WMMA_F32_16X16X4_F32.
// ---------------------------------------------------------------------------
__global__ void attn_kernel(const float* __restrict__ x,
                            const float* __restrict__ sim,
                            const float* __restrict__ thr,
                            const float* __restrict__ isum,
                            float* __restrict__ out1) {
  const int lane = threadIdx.x & 31;
  const int wid  = blockIdx.x * (blockDim.x >> 5) + (threadIdx.x >> 5);
  // per batch: (T/32)=64 t-tiles x (C/64)=4 c-tiles
  const int b   = wid / 256;
  const int rem = wid % 256;
  const int t_base = (rem / 4) * 32;
  const int c_base = (rem % 4) * 64;

  const float* simb = sim + (size_t)b * T_ * T_;
  const float* xb   = x + (size_t)b * C_ * T_;

  const int row = lane & 15;
  const int kk  = (lane >> 4) * 2;

  float th[2], is[2];
#pragma unroll
  for (int mt = 0; mt < 2; ++mt) {
    const int t = t_base + mt * 16 + row;
    th[mt] = thr[b * T_ + t];
    is[mt] = isum[b * T_ + t];
  }

  v8f acc[2][4] = {};

  for (int s0 = 0; s0 < T_; s0 += 4) {
    v2f a[2], bb[4];
#pragma unroll
    for (int mt = 0; mt < 2; ++mt) {
      const float* p = simb + (size_t)(t_base + mt * 16 + row) * T_ + s0 + kk;
      const float v0 = p[0], v1 = p[1];          // contiguous pair
      a[mt].x = (v0 >= th[mt]) ? v0 * is[mt] : 0.f;
      a[mt].y = (v1 >= th[mt]) ? v1 * is[mt] : 0.f;
    }
#pragma unroll
    for (int nt = 0; nt < 4; ++nt) {
      const float* p = xb + (size_t)(c_base + nt * 16 + row) * T_ + s0 + kk;
      bb[nt].x = p[0];                            // v[s,c] = x[b,c,s]
      bb[nt].y = p[1];
    }
#pragma unroll
    for (int mt = 0; mt < 2; ++mt)
#pragma unroll
      for (int nt = 0; nt < 4; ++nt)
        acc[mt][nt] = WMMA_F32(a[mt], bb[nt], acc[mt][nt]);
  }

#pragma unroll
  for (int mt = 0; mt < 2; ++mt)
#pragma unroll
    for (int nt = 0; nt < 4; ++nt) {
      const int c = c_base + nt * 16 + row;
#pragma unroll
      for (int i = 0; i < 8; ++i) {
        const int t = t_base + mt * 16 + i + (lane >> 4) * 8;
        const size_t idx = (size_t)b * C_ * T_ + (size_t)c * T_ + t;
        out1[idx] = acc[mt][nt][i] + x[idx];      // +v residual, transposed store
      }
    }
}

// ---------------------------------------------------------------------------
extern "C" void kernel_launch(void* const* d_in, const int* in_sizes, int n_in,
                              void* d_out, int out_size, void* d_ws, size_t ws_size,
                              hipStream_t stream) {
  (void)in_sizes; (void)n_in; (void)out_size; (void)ws_size;
  const float* x = (const float*)d_in[0];

  float* out  = (float*)d_out;
  float* out1 = out;                                    // (16,256,2048)
  float* sim  = out + (size_t)B_ * C_ * T_;             // (16,2048,2048)

  float* invn = (float*)d_ws;                           // B*T floats
  float* thr  = invn + B_ * T_;                         // B*T floats
  float* isum = thr  + B_ * T_;                         // B*T floats

  // 1) inverse channel norms
  norm_kernel<<<(B_ * T_) / 256, 256, 0, stream>>>(x, invn);
  // 2) cosine-similarity Gram matrix (fp32 WMMA): 32768 waves, 4 waves/WG
  gram_kernel<<<8192, 128, 0, stream>>>(x, invn, sim);
  // 3) exact top-k threshold + row sums: one block per row
  topk_kernel<<<B_ * T_, 256, 0, stream>>>(sim, thr, isum);
  // 4) graph @ v + v (fp32 WMMA), transposed store: 4096 waves, 4 waves/WG
  attn_kernel<<<1024, 128, 0, stream>>>(x, sim, thr, isum, out1);
}